// ModelAndLoss_75153337745702
// MI455X (gfx1250) — compile-verified
//
#include <hip/hip_runtime.h>
#include <math.h>

// ---------------------------------------------------------------------------
// Problem constants
// ---------------------------------------------------------------------------
#define NROW 64            // N
#define NCLS 1000          // C
#define NFA  3             // features in group A
#define NFB  4             // features in group B
#define NF   (NFA + NFB)   // 7 total features
#define DA   65536
#define DB   32768
#define NRT  4             // row tiles (16 rows each) covering 64 rows
#define NCT  4             // column tiles per wave (full 64 cols)
#define NSEG 32            // K-dim segments per (feature,row-tile)
#define TILE_ELEMS 256     // 16x16 floats
#define STRIP_ELEMS (NCT * TILE_ELEMS)   // 1024 floats per wave

#define NWAVE (NF * NRT * NSEG)                     // 896 gram waves
#define PART_FLOATS ((size_t)NWAVE * STRIP_ELEMS)   // 917504
#define K_FLOATS    ((size_t)NF * NROW * NROW)      // 28672

typedef __attribute__((ext_vector_type(2))) float v2f;
typedef __attribute__((ext_vector_type(8))) float v8f;

// ---------------------------------------------------------------------------
// 1) Cross-entropy: one block per row. rowloss[row] = -logp[row, target[row]]
// ---------------------------------------------------------------------------
__global__ void ce_kernel(const float* __restrict__ output,
                          const int* __restrict__ target,
                          float* __restrict__ rowloss) {
    const int row = blockIdx.x;
    const int t = threadIdx.x;           // 128 threads
    __shared__ float sh[128];
    const float* x = output + row * NCLS;

    float mx = -INFINITY;
    for (int c = t; c < NCLS; c += 128) mx = fmaxf(mx, x[c]);
    sh[t] = mx; __syncthreads();
    for (int s = 64; s > 0; s >>= 1) {
        if (t < s) sh[t] = fmaxf(sh[t], sh[t + s]);
        __syncthreads();
    }
    const float m = sh[0]; __syncthreads();

    float se = 0.f;
    for (int c = t; c < NCLS; c += 128) se += expf(x[c] - m);
    sh[t] = se; __syncthreads();
    for (int s = 64; s > 0; s >>= 1) {
        if (t < s) sh[t] += sh[t + s];
        __syncthreads();
    }
    if (t == 0) {
        const int tg = target[row];
        rowloss[row] = -(x[tg] - m - logf(sh[0]));
    }
}

// ---------------------------------------------------------------------------
// 2) Gram via WMMA: one wave per (feature, row-tile, K-segment) computes a
//    full 16x64 strip (4 column tiles). K_f = X_f X_f^T means the B operand
//    (B = A^T, 4x16 K-major) has the *same* lane layout as A (16x4), so each
//    lane loads one float2 per row-tile. A is loaded ONCE per K-step and
//    reused across 4 WMMAs into 4 independent accumulators (5 loads : 4 WMMA,
//    plus 4-way WMMA ILP per wave).
//    A 16x4 f32 layout: lane l holds M=l&15, K = 2*(l>>4) + {0,1}.
// ---------------------------------------------------------------------------
__global__ void gram_kernel(const float* __restrict__ fa,
                            const float* __restrict__ fb,
                            float* __restrict__ part) {
    const int bid = blockIdx.x;                 // NF*NRT*NSEG waves
    const int f   = bid / (NRT * NSEG);
    const int rem = bid % (NRT * NSEG);
    const int ti  = rem / NSEG;                 // row tile 0..3
    const int seg = rem % NSEG;

    const float* X;
    int D;
    if (f < NFA) { X = fa + (size_t)f * NROW * DA;          D = DA; }
    else         { X = fb + (size_t)(f - NFA) * NROW * DB;  D = DB; }

    const int lane = threadIdx.x;               // wave32
    const int m  = lane & 15;
    const int kh = (lane >> 4) << 1;            // 0 or 2

    const float* arow = X + (size_t)(ti * 16 + m) * D + kh;
    const float* br0  = X + (size_t)m * D + kh; // column tile 0 (rows 0..15)
    const float* br1  = br0 + (size_t)16 * D;
    const float* br2  = br0 + (size_t)32 * D;
    const float* br3  = br0 + (size_t)48 * D;

    const int seglen = D >> 5;                  // D / NSEG
    const int kbeg = seg * seglen;
    const int kend = kbeg + seglen;

    v8f c0 = {}, c1 = {}, c2 = {}, c3 = {};
    #pragma unroll 4
    for (int k = kbeg; k < kend; k += 4) {
        v2f a  = *(const v2f*)(arow + k);
        v2f q0 = *(const v2f*)(br0 + k);
        v2f q1 = *(const v2f*)(br1 + k);
        v2f q2 = *(const v2f*)(br2 + k);
        v2f q3 = *(const v2f*)(br3 + k);
        c0 = __builtin_amdgcn_wmma_f32_16x16x4_f32(false, a, false, q0,
                                                   (short)0, c0, false, false);
        c1 = __builtin_amdgcn_wmma_f32_16x16x4_f32(false, a, false, q1,
                                                   (short)0, c1, false, false);
        c2 = __builtin_amdgcn_wmma_f32_16x16x4_f32(false, a, false, q2,
                                                   (short)0, c2, false, false);
        c3 = __builtin_amdgcn_wmma_f32_16x16x4_f32(false, a, false, q3,
                                                   (short)0, c3, false, false);
    }

    // Strip layout: part[bid][tj][r*32 + lane]
    float* p = part + (size_t)bid * STRIP_ELEMS;
    #pragma unroll
    for (int r = 0; r < 8; ++r) {
        p[0 * TILE_ELEMS + r * 32 + lane] = c0[r];
        p[1 * TILE_ELEMS + r * 32 + lane] = c1[r];
        p[2 * TILE_ELEMS + r * 32 + lane] = c2[r];
        p[3 * TILE_ELEMS + r * 32 + lane] = c3[r];
    }
}

// ---------------------------------------------------------------------------
// 3) Reduce 32 segment partials -> K[f][64][64], zero diagonal.
//    C/D layout: VGPR r, lanes 0-15 -> M=r, N=lane ; lanes 16-31 -> M=r+8.
// ---------------------------------------------------------------------------
__global__ void reduce_kernel(const float* __restrict__ part,
                              float* __restrict__ K) {
    const int b = blockIdx.x;                   // NF * 16 tiles
    const int f = b / 16;
    const int tile = b % 16;
    const int ti = tile >> 2, tj = tile & 3;
    const int e = threadIdx.x;                  // 256

    // partial base for (f, ti), column tile tj, element e; stride per segment
    // is one full strip (1024 floats).
    const float* p = part + ((size_t)(f * NRT + ti) * NSEG) * STRIP_ELEMS
                          + (size_t)tj * TILE_ELEMS + e;
    float acc = 0.f;
    #pragma unroll 4
    for (int s = 0; s < NSEG; ++s) acc += p[(size_t)s * STRIP_ELEMS];

    const int r = e >> 5;
    const int lane = e & 31;
    const int M = r + ((lane >> 4) << 3);
    const int Nc = lane & 15;
    const int gm = ti * 16 + M;
    const int gn = tj * 16 + Nc;
    K[(size_t)f * (NROW * NROW) + gm * NROW + gn] = (gm == gn) ? 0.f : acc;
}

// ---------------------------------------------------------------------------
// 4) CKA statistics + final loss (double precision, matches f64 reference).
//    hsic[i,j] = (tr(KiKj) + s_i s_j/((N-1)(N-2)) - 2<r_i,r_j>/(N-2)) / (N(N-3))
//    cka sum = sum_{same group pairs} hsic[i,j]/(d_i d_j),  d_i = sqrt(hsic_ii)
// ---------------------------------------------------------------------------
__global__ void stats_kernel(const float* __restrict__ K,
                             const float* __restrict__ rowloss,
                             float* __restrict__ out0) {
    const int t = threadIdx.x;                  // 256 threads
    __shared__ double sh[256];
    __shared__ double trKL[NF * NF];
    __shared__ double rsum[NF][NROW];
    __shared__ double ssum[NF];

    // row sums r[f][m]
    for (int idx = t; idx < NF * NROW; idx += 256) {
        const int f = idx / NROW, m = idx % NROW;
        double a = 0.0;
        const float* kr = K + (size_t)f * (NROW * NROW) + m * NROW;
        for (int n = 0; n < NROW; ++n) a += (double)kr[n];
        rsum[f][m] = a;
    }
    __syncthreads();
    if (t < NF) {
        double a = 0.0;
        for (int m = 0; m < NROW; ++m) a += rsum[t][m];
        ssum[t] = a;
    }
    __syncthreads();

    // tr(Ki Kj) = elementwise <Ki, Kj> (K symmetric); only same-group pairs
    for (int p = 0; p < NF * NF; ++p) {
        const int i = p / NF, j = p % NF;
        const bool same = (i < NFA) == (j < NFA);
        double a = 0.0;
        if (same) {
            const float* Ki = K + (size_t)i * (NROW * NROW);
            const float* Kj = K + (size_t)j * (NROW * NROW);
            for (int e = t; e < NROW * NROW; e += 256)
                a += (double)Ki[e] * (double)Kj[e];
        }
        sh[t] = a; __syncthreads();
        for (int s = 128; s > 0; s >>= 1) {
            if (t < s) sh[t] += sh[t + s];
            __syncthreads();
        }
        if (t == 0) trKL[p] = sh[0];
        __syncthreads();
    }

    if (t == 0) {
        const double c1 = 1.0 / ((NROW - 1.0) * (NROW - 2.0));   // 1/(63*62)
        const double c2 = 2.0 / (NROW - 2.0);                    // 2/62
        const double c3 = 1.0 / ((double)NROW * (NROW - 3.0));   // 1/(64*61)
        double hs[NF * NF];
        for (int i = 0; i < NF; ++i)
            for (int j = 0; j < NF; ++j) {
                if ((i < NFA) != (j < NFA)) { hs[i * NF + j] = 0.0; continue; }
                double rr = 0.0;
                for (int m = 0; m < NROW; ++m) rr += rsum[i][m] * rsum[j][m];
                hs[i * NF + j] =
                    (trKL[i * NF + j] + ssum[i] * ssum[j] * c1 - c2 * rr) * c3;
            }
        double d[NF];
        for (int i = 0; i < NF; ++i) d[i] = sqrt(hs[i * NF + i]);
        double cka = 0.0;
        for (int i = 0; i < NF; ++i)
            for (int j = 0; j < NF; ++j) {
                if ((i < NFA) != (j < NFA)) continue;
                cka += hs[i * NF + j] / (d[i] * d[j]);
            }
        double ce = 0.0;
        for (int m = 0; m < NROW; ++m) ce += (double)rowloss[m];
        ce /= (double)NROW;
        out0[0] = (float)(ce + 0.1 * cka);
    }
}

// ---------------------------------------------------------------------------
// 5) Copy logits through to d_out[1:] (tuple return: (loss, output))
// ---------------------------------------------------------------------------
__global__ void copy_out_kernel(const float* __restrict__ src,
                                float* __restrict__ dst, int n) {
    int i = blockIdx.x * blockDim.x + threadIdx.x;
    if (i < n) dst[i] = src[i];
}

// ---------------------------------------------------------------------------
// Launch
// ---------------------------------------------------------------------------
extern "C" void kernel_launch(void* const* d_in, const int* in_sizes, int n_in,
                              void* d_out, int out_size, void* d_ws, size_t ws_size,
                              hipStream_t stream) {
    const float* output = (const float*)d_in[0];   // [64,1000] f32
    const int*   target = (const int*)d_in[1];     // [64] int
    const float* fa     = (const float*)d_in[2];   // [3,64,65536] f32
    const float* fb     = (const float*)d_in[3];   // [4,64,32768] f32
    float* out = (float*)d_out;                    // [1 + 64000] f32

    float* ws      = (float*)d_ws;
    float* part    = ws;                           // PART_FLOATS
    float* Kmat    = ws + PART_FLOATS;             // K_FLOATS
    float* rowloss = Kmat + K_FLOATS;              // 64

    ce_kernel<<<NROW, 128, 0, stream>>>(output, target, rowloss);
    gram_kernel<<<NWAVE, 32, 0, stream>>>(fa, fb, part);
    reduce_kernel<<<NF * 16, 256, 0, stream>>>(part, Kmat);
    stats_kernel<<<1, 256, 0, stream>>>(Kmat, rowloss, out);
    copy_out_kernel<<<(NROW * NCLS + 255) / 256, 256, 0, stream>>>(
        output, out + 1, NROW * NCLS);
}